// BCE_topK_loss_59339268162115
// MI455X (gfx1250) — compile-verified
//
#include <hip/hip_runtime.h>
#include <cstdint>
#include <cstddef>

#define DEV __device__ __forceinline__

typedef float    f4 __attribute__((ext_vector_type(4)));
typedef unsigned u4 __attribute__((ext_vector_type(4)));

constexpr unsigned SLICE_N = 12582912u;  // 192*256*256 spatial elements per (B,C) slice
constexpr unsigned NTOP    = 1258291u;   // round(SLICE_N * 10 / 100)
constexpr int      NSLICE  = 2;          // B*C = 2*1

// ---------------- workspace layout (bytes, d_ws is 256B aligned) ----------------
constexpr size_t OFF_SUMABOVE = 0;                              // double[2]
constexpr size_t OFF_BINSUM1  = 16;                             // double[2][4096]
constexpr size_t OFF_BINSUM2  = OFF_BINSUM1 + (size_t)2*4096*8; // double[2][256]
constexpr size_t OFF_HIST0    = OFF_BINSUM2 + (size_t)2*256*8;  // uint[2][4096]
constexpr size_t OFF_HIST1    = OFF_HIST0   + (size_t)2*4096*4; // uint[2][4096]
constexpr size_t OFF_HIST2    = OFF_HIST1   + (size_t)2*4096*4; // uint[2][256]
constexpr size_t OFF_SEL      = OFF_HIST2   + (size_t)2*256*4;  // uint[8]: b0[2],r0[2],b1[2],r1[2]
constexpr size_t AUX_ZERO_BYTES = OFF_SEL + 8*4;
constexpr size_t OFF_BITS     = (AUX_ZERO_BYTES + 255) & ~(size_t)255; // uint[2*SLICE_N]

constexpr int TPB    = 256;   // 8 wave32s per block
constexpr int GRID_X = 1024;

// ---------------- numerically stable BCE-with-logits (reduction='none') ----------------
DEV float bce(float x, float t) {
    // max(x,0) - x*t + log1p(exp(-|x|))  > 0 always  -> uint bit pattern orders like value
    return fmaxf(x, 0.0f) - x * t + log1pf(expf(-fabsf(x)));
}

// ---------------- gfx1250 async global->LDS helpers ----------------
DEV unsigned lds_off(const void* p) {
    // LDS aperture lives in bits[63:32] of the generic address; low 32 bits are the
    // LDS byte offset (ISA 10.2 aperture mapping).
    return (unsigned)(size_t)p;
}

DEV void async_cp_b128(const unsigned* gsrc, unsigned lds_byte_off) {
    asm volatile("global_load_async_to_lds_b128 %0, %1, off"
                 :: "v"(lds_byte_off), "v"(gsrc)
                 : "memory");
}

// Always inline asm with a "memory" clobber: this must be a *compiler* barrier too,
// so the subsequent ds_load of the staging buffer cannot be scheduled before the wait.
template <int N>
DEV void wait_async() {
    asm volatile("s_wait_asynccnt %0" :: "n"(N) : "memory");
}

// =======================================================================================
// Pass 1: compute loss, (optionally) store its bit pattern, histogram bits[31:20]
// grid = (GRID_X, NSLICE), block = TPB
// =======================================================================================
template <bool STORE>
__global__ __launch_bounds__(TPB)
void k_pass1(const float* __restrict__ x, const float* __restrict__ t,
             unsigned* __restrict__ bits, unsigned* __restrict__ hist0) {
    __shared__ unsigned lh[4096];
    const unsigned s = blockIdx.y;
    for (unsigned i = threadIdx.x; i < 4096; i += blockDim.x) lh[i] = 0;
    __syncthreads();

    const size_t   base = (size_t)s * SLICE_N;
    const unsigned nthr = gridDim.x * blockDim.x;
    for (unsigned i4 = blockIdx.x * blockDim.x + threadIdx.x; i4 < SLICE_N / 4; i4 += nthr) {
        const size_t idx = base + (size_t)i4 * 4;
        // NT loads: inputs are streamed exactly once -> keep them out of L2 so the
        // bits array (100.7 MB < 192 MB L2) stays resident for passes 2 and 3.
        const f4 xv = __builtin_nontemporal_load((const f4*)(x + idx));
        const f4 tv = __builtin_nontemporal_load((const f4*)(t + idx));
        const unsigned b0 = __float_as_uint(bce(xv.x, tv.x));
        const unsigned b1 = __float_as_uint(bce(xv.y, tv.y));
        const unsigned b2 = __float_as_uint(bce(xv.z, tv.z));
        const unsigned b3 = __float_as_uint(bce(xv.w, tv.w));
        atomicAdd(&lh[b0 >> 20], 1u);
        atomicAdd(&lh[b1 >> 20], 1u);
        atomicAdd(&lh[b2 >> 20], 1u);
        atomicAdd(&lh[b3 >> 20], 1u);
        if constexpr (STORE) {
            u4 v; v.x = b0; v.y = b1; v.z = b2; v.w = b3;
            *(u4*)(bits + idx) = v;
        }
    }
    __syncthreads();
    for (unsigned i = threadIdx.x; i < 4096; i += blockDim.x) {
        const unsigned c = lh[i];
        if (c) atomicAdd(&hist0[s * 4096 + i], c);
    }
}

// =======================================================================================
// Level-0 selection: per slice, find bin b0 containing the NTOP-th largest, r0 = remaining
// =======================================================================================
__global__ void k_sel0(const unsigned* __restrict__ hist0, unsigned* __restrict__ sel) {
    const unsigned s = threadIdx.x;
    if (s >= NSLICE) return;
    unsigned cum = 0;
    int b = 4095;
    for (; b > 0; --b) {
        const unsigned c = hist0[s * 4096 + b];
        if (cum + c >= NTOP) break;
        cum += c;
    }
    sel[0 + s] = (unsigned)b;   // b0
    sel[2 + s] = NTOP - cum;    // r0  (1..hist0[b0])
}

// =======================================================================================
// Refinement pass (level 1 / level 2  x  bits-path / recompute-path).
//  L3=false: key = b0 (bits>>20), bin = bits[19:8] (4096 bins), also sum everything above.
//  L3=true : key = (b0<<12)|b1 (bits>>8), bin = bits[7:0] (256 bins).
// FROM_BITS=true streams the stored loss bits through LDS with double-buffered
// global_load_async_to_lds_b128 (per-wave 512B tiles, ASYNCcnt-pipelined; bits are
// L2-resident so this pass runs at L2 bandwidth).
// =======================================================================================
template <bool L3, bool FROM_BITS>
__global__ __launch_bounds__(TPB)
void k_refine(const float* __restrict__ x, const float* __restrict__ t,
              const unsigned* __restrict__ bits, const unsigned* __restrict__ sel,
              unsigned* __restrict__ hist, double* __restrict__ binSum,
              double* __restrict__ sumAbove) {
    constexpr int BINS = L3 ? 256 : 4096;
    __shared__ unsigned lh[BINS];
    __shared__ double   ls[BINS];
    __shared__ double   red[TPB];
    __shared__ __align__(16) unsigned stage[2][TPB / 32][128];

    const unsigned s   = blockIdx.y;
    const unsigned key = L3 ? ((sel[0 + s] << 12) | sel[4 + s]) : sel[0 + s];
    for (unsigned i = threadIdx.x; i < (unsigned)BINS; i += blockDim.x) { lh[i] = 0; ls[i] = 0.0; }
    __syncthreads();

    const size_t base = (size_t)s * SLICE_N;
    double acc = 0.0;

    auto proc = [&](unsigned b) {
        const unsigned top = L3 ? (b >> 8) : (b >> 20);
        if (!L3 && top > key) {
            acc += (double)__uint_as_float(b);
        } else if (top == key) {
            const unsigned bin = L3 ? (b & 0xFFu) : ((b >> 8) & 0xFFFu);
            atomicAdd(&lh[bin], 1u);
            atomicAdd(&ls[bin], (double)__uint_as_float(b));
        }
    };

    if constexpr (FROM_BITS) {
        constexpr unsigned WPB = TPB / 32;
        const unsigned wave = threadIdx.x >> 5, lane = threadIdx.x & 31;
        const unsigned nch = SLICE_N / 128;           // 512B chunks per slice
        const unsigned wstride = gridDim.x * WPB;
        unsigned chunk = blockIdx.x * WPB + wave;     // wave-uniform
        int buf = 0;
        if (chunk < nch)
            async_cp_b128(bits + base + (size_t)chunk * 128 + lane * 4,
                          lds_off(&stage[0][wave][lane * 4]));
        while (chunk < nch) {
            const unsigned nxt = chunk + wstride;
            if (nxt < nch) {
                async_cp_b128(bits + base + (size_t)nxt * 128 + lane * 4,
                              lds_off(&stage[buf ^ 1][wave][lane * 4]));
                wait_async<1>();   // oldest (current buf) complete; next stays in flight
            } else {
                wait_async<0>();
            }
            const u4 v = *(const u4*)&stage[buf][wave][lane * 4];
            proc(v.x); proc(v.y); proc(v.z); proc(v.w);
            buf ^= 1;
            chunk = nxt;
        }
    } else {
        const unsigned nthr = gridDim.x * blockDim.x;
        for (unsigned i4 = blockIdx.x * blockDim.x + threadIdx.x; i4 < SLICE_N / 4; i4 += nthr) {
            const size_t idx = base + (size_t)i4 * 4;
            const f4 xv = *(const f4*)(x + idx);
            const f4 tv = *(const f4*)(t + idx);
            proc(__float_as_uint(bce(xv.x, tv.x)));
            proc(__float_as_uint(bce(xv.y, tv.y)));
            proc(__float_as_uint(bce(xv.z, tv.z)));
            proc(__float_as_uint(bce(xv.w, tv.w)));
        }
    }

    if constexpr (!L3) {
        red[threadIdx.x] = acc;
        __syncthreads();
        for (int st = TPB / 2; st > 0; st >>= 1) {
            if ((int)threadIdx.x < st) red[threadIdx.x] += red[threadIdx.x + st];
            __syncthreads();
        }
        if (threadIdx.x == 0) atomicAdd(&sumAbove[s], red[0]);
    } else {
        __syncthreads();
    }

    for (unsigned i = threadIdx.x; i < (unsigned)BINS; i += blockDim.x) {
        const unsigned c = lh[i];
        if (c) atomicAdd(&hist[s * BINS + i], c);
        const double v = ls[i];
        if (v != 0.0) atomicAdd(&binSum[s * BINS + i], v);
    }
}

// =======================================================================================
// Level-1 selection: find b1, r1; fold sum of mid-bins strictly above b1 into sumAbove.
// =======================================================================================
__global__ void k_sel1(const unsigned* __restrict__ hist1, const double* __restrict__ binSum1,
                       unsigned* __restrict__ sel, double* __restrict__ sumAbove) {
    const unsigned s = threadIdx.x;
    if (s >= NSLICE) return;
    const unsigned r0 = sel[2 + s];
    unsigned cum = 0;
    double sa = 0.0;
    int b = 4095;
    for (; b > 0; --b) {
        const unsigned c = hist1[s * 4096 + b];
        if (cum + c >= r0) break;
        cum += c;
        sa += binSum1[s * 4096 + b];
    }
    sel[4 + s] = (unsigned)b;
    sel[6 + s] = r0 - cum;
    sumAbove[s] += sa;   // single writer at this point in the stream
}

// =======================================================================================
// Final: exact threshold tau, Sum(top-n) = Sum(>tau) + r*tau (ties), mean over slices.
// =======================================================================================
__global__ void k_final(const unsigned* __restrict__ hist2, const double* __restrict__ binSum2,
                        const double* __restrict__ sumAbove, const unsigned* __restrict__ sel,
                        float* __restrict__ out) {
    double total = 0.0;
    for (unsigned s = 0; s < NSLICE; ++s) {
        const unsigned r1 = sel[6 + s];
        const unsigned b0 = sel[0 + s], b1 = sel[4 + s];
        unsigned cum = 0;
        double sa = 0.0;
        int b = 255;
        for (; b > 0; --b) {
            const unsigned c = hist2[s * 256 + b];
            if (cum + c >= r1) break;
            cum += c;
            sa += binSum2[s * 256 + b];
        }
        const unsigned r2  = r1 - cum;                       // ties taken exactly at tau
        const unsigned tau = (b0 << 20) | (b1 << 8) | (unsigned)b;
        const double sum_top = sumAbove[s] + sa + (double)r2 * (double)__uint_as_float(tau);
        total += sum_top / (double)NTOP;
    }
    out[0] = (float)(total / (double)NSLICE);
}

// =======================================================================================
extern "C" void kernel_launch(void* const* d_in, const int* in_sizes, int n_in,
                              void* d_out, int out_size, void* d_ws, size_t ws_size,
                              hipStream_t stream) {
    (void)in_sizes; (void)n_in; (void)out_size;
    const float* x = (const float*)d_in[0];   // net_output (logits)
    const float* t = (const float*)d_in[1];   // target

    char*     ws       = (char*)d_ws;
    double*   sumAbove = (double*)(ws + OFF_SUMABOVE);
    double*   binSum1  = (double*)(ws + OFF_BINSUM1);
    double*   binSum2  = (double*)(ws + OFF_BINSUM2);
    unsigned* hist0    = (unsigned*)(ws + OFF_HIST0);
    unsigned* hist1    = (unsigned*)(ws + OFF_HIST1);
    unsigned* hist2    = (unsigned*)(ws + OFF_HIST2);
    unsigned* sel      = (unsigned*)(ws + OFF_SEL);
    unsigned* bits     = (unsigned*)(ws + OFF_BITS);

    const bool useBits = ws_size >= OFF_BITS + (size_t)NSLICE * SLICE_N * 4;

    hipMemsetAsync(d_ws, 0, AUX_ZERO_BYTES, stream);  // zero all accumulators every call

    const dim3 grid(GRID_X, NSLICE, 1);
    if (useBits) {
        k_pass1<true ><<<grid, TPB, 0, stream>>>(x, t, bits, hist0);
    } else {
        k_pass1<false><<<grid, TPB, 0, stream>>>(x, t, nullptr, hist0);
    }
    k_sel0<<<1, 32, 0, stream>>>(hist0, sel);

    if (useBits) {
        k_refine<false, true ><<<grid, TPB, 0, stream>>>(x, t, bits, sel, hist1, binSum1, sumAbove);
    } else {
        k_refine<false, false><<<grid, TPB, 0, stream>>>(x, t, nullptr, sel, hist1, binSum1, sumAbove);
    }
    k_sel1<<<1, 32, 0, stream>>>(hist1, binSum1, sel, sumAbove);

    if (useBits) {
        k_refine<true, true ><<<grid, TPB, 0, stream>>>(x, t, bits, sel, hist2, binSum2, nullptr);
    } else {
        k_refine<true, false><<<grid, TPB, 0, stream>>>(x, t, nullptr, sel, hist2, binSum2, nullptr);
    }
    k_final<<<1, 1, 0, stream>>>(hist2, binSum2, sumAbove, sel, (float*)d_out);
}